// GCN2_85031762526673
// MI455X (gfx1250) — compile-verified
//
#include <hip/hip_runtime.h>

typedef __attribute__((ext_vector_type(2))) float v2f;
typedef __attribute__((ext_vector_type(8))) float v8f;

#define C 64

// ---------------- init kernels ----------------
__global__ void init_deg_kernel(float* __restrict__ deg, int N) {
    int i = blockIdx.x * 256 + threadIdx.x;
    if (i < N) deg[i] = 1.0f;          // self-loop weight
}

__global__ void zero_agg_kernel(float* __restrict__ agg, int n) {
    int i = blockIdx.x * 256 + threadIdx.x;
    if (i < n) agg[i] = 0.0f;
}

// ---------------- degree accumulation ----------------
__global__ void accum_deg_kernel(const long long* __restrict__ col,
                                 const float* __restrict__ w,
                                 float* __restrict__ deg, int E) {
    int e = blockIdx.x * 256 + threadIdx.x;
    if (e < E) atomicAdd(&deg[(int)col[e]], w[e]);
}

__global__ void dinv_kernel(const float* __restrict__ deg,
                            float* __restrict__ dinv, int N) {
    int i = blockIdx.x * 256 + threadIdx.x;
    if (i < N) {
        float d = deg[i];
        dinv[i] = (d > 0.0f) ? rsqrtf(d) : 0.0f;
    }
}

// ---------------- edge scatter: one wave32 per edge ----------------
// Edge metadata is wave-uniform -> force scalar loads via readfirstlane.
__global__ __launch_bounds__(256) void scatter_kernel(
    const float* __restrict__ x,
    const long long* __restrict__ row_idx,
    const long long* __restrict__ col_idx,
    const float* __restrict__ w,
    const float* __restrict__ dinv,
    float* __restrict__ agg, int E) {
    int lane = threadIdx.x & 31;
    int gw = __builtin_amdgcn_readfirstlane(blockIdx.x * 8 + (threadIdx.x >> 5));
    if (gw >= E) return;
    int r = (int)row_idx[gw];          // s_load_b64
    int c = (int)col_idx[gw];          // s_load_b64
    float norm = dinv[r] * w[gw] * dinv[c];   // s_load_b32 x3
    float2 v = ((const float2*)(x + (size_t)c * C))[lane];  // vector gather (L2-resident)
    float* dst = agg + (size_t)r * C + lane * 2;
    atomicAdd(dst + 0, v.x * norm);
    atomicAdd(dst + 1, v.y * norm);
}

// ---------------- fused residual + GEMM(relu) with f32 WMMA ----------------
// h = 0.9*(agg + dinv^2 * x) + 0.1*x ; out = relu(h @ W)
// 128 threads = 4 waves; each wave computes a 32x64 output block
// (two 16-row tiles sharing B fragments) via V_WMMA_F32_16X16X4_F32.
__global__ __launch_bounds__(128) void gemm_relu_kernel(
    const float* __restrict__ x,
    const float* __restrict__ agg,
    const float* __restrict__ dinv,
    const float* __restrict__ W,
    float* __restrict__ out, int N) {
    // sB[kp][n] = {W[2kp][n], W[2kp+1][n]} : one ds_load_b64 per B fragment.
    // float2 stride 80 => 160 floats == 32 mod 64 -> lane-halves hit disjoint banks.
    __shared__ float2 sB[C / 2][80];
    // stride 68 floats == 4 mod 64 -> conflict-free b64 A-frag reads.
    __shared__ float sH[4][32][68];

    int tid = threadIdx.x;

    // stage W into paired layout: item = (kp, n4) -> 2 global float4 + 4 ds_store_b64
    const float4* W4 = (const float4*)W;
    for (int i = tid; i < (C / 2) * (C / 4); i += 128) {
        int kp = i >> 4, n4 = i & 15;
        float4 wa = W4[(2 * kp + 0) * 16 + n4];
        float4 wb = W4[(2 * kp + 1) * 16 + n4];
        sB[kp][n4 * 4 + 0] = make_float2(wa.x, wb.x);
        sB[kp][n4 * 4 + 1] = make_float2(wa.y, wb.y);
        sB[kp][n4 * 4 + 2] = make_float2(wa.z, wb.z);
        sB[kp][n4 * 4 + 3] = make_float2(wa.w, wb.w);
    }

    int wave = tid >> 5;
    int lane = tid & 31;
    int row0 = (blockIdx.x * 4 + wave) * 32;
    bool active = (row0 < N);            // wave-uniform (N % 32 == 0)

    if (active) {
        // stage h block (32x64) into LDS: coalesced float4 loads of agg and x
        for (int i = lane; i < 32 * 16; i += 32) {
            int rr = i >> 4, c4 = i & 15;
            size_t base = (size_t)(row0 + rr) * C + c4 * 4;
            float4 a  = *(const float4*)(agg + base);
            float4 xv = *(const float4*)(x + base);
            float di = dinv[row0 + rr];
            float sn = di * di;          // self-loop norm dinv*1*dinv
            float4 h;
            h.x = 0.9f * (a.x + sn * xv.x) + 0.1f * xv.x;
            h.y = 0.9f * (a.y + sn * xv.y) + 0.1f * xv.y;
            h.z = 0.9f * (a.z + sn * xv.z) + 0.1f * xv.z;
            h.w = 0.9f * (a.w + sn * xv.w) + 0.1f * xv.w;
            *(float4*)&sH[wave][rr][c4 * 4] = h;
        }
    }
    __syncthreads();
    if (!active) return;                 // whole wave exits -> EXEC all-1s for WMMA

    int laneM = lane & 15;               // M (A) / N (B,C,D) position
    int kh    = lane >> 4;               // lane-half

    v8f acc0 = {}, acc1 = {}, acc2 = {}, acc3 = {};
    v8f acc4 = {}, acc5 = {}, acc6 = {}, acc7 = {};

#pragma unroll
    for (int k0 = 0; k0 < C; k0 += 4) {
        int ka = k0 + 2 * kh;            // A: lane-half selects K pair
        int kp = (k0 >> 1) + kh;         // B: paired-K row in sB
        v2f a0 = *(const v2f*)&sH[wave][laneM][ka];
        v2f a1 = *(const v2f*)&sH[wave][16 + laneM][ka];
        v2f b0 = *(const v2f*)&sB[kp][laneM +  0];
        v2f b1 = *(const v2f*)&sB[kp][laneM + 16];
        v2f b2 = *(const v2f*)&sB[kp][laneM + 32];
        v2f b3 = *(const v2f*)&sB[kp][laneM + 48];
        acc0 = __builtin_amdgcn_wmma_f32_16x16x4_f32(false, a0, false, b0, (short)0, acc0, false, false);
        acc1 = __builtin_amdgcn_wmma_f32_16x16x4_f32(false, a0, false, b1, (short)0, acc1, false, false);
        acc2 = __builtin_amdgcn_wmma_f32_16x16x4_f32(false, a0, false, b2, (short)0, acc2, false, false);
        acc3 = __builtin_amdgcn_wmma_f32_16x16x4_f32(false, a0, false, b3, (short)0, acc3, false, false);
        acc4 = __builtin_amdgcn_wmma_f32_16x16x4_f32(false, a1, false, b0, (short)0, acc4, false, false);
        acc5 = __builtin_amdgcn_wmma_f32_16x16x4_f32(false, a1, false, b1, (short)0, acc5, false, false);
        acc6 = __builtin_amdgcn_wmma_f32_16x16x4_f32(false, a1, false, b2, (short)0, acc6, false, false);
        acc7 = __builtin_amdgcn_wmma_f32_16x16x4_f32(false, a1, false, b3, (short)0, acc7, false, false);
    }

    // C/D layout: VGPR v -> row v (lanes 0-15) / v+8 (lanes 16-31), col = laneM
#pragma unroll
    for (int v = 0; v < 8; v++) {
        int rr = v + 8 * kh;
        float* o0 = out + (size_t)(row0 + rr) * C + laneM;
        o0[0]  = fmaxf(acc0[v], 0.0f);
        o0[16] = fmaxf(acc1[v], 0.0f);
        o0[32] = fmaxf(acc2[v], 0.0f);
        o0[48] = fmaxf(acc3[v], 0.0f);
        float* o1 = out + (size_t)(row0 + 16 + rr) * C + laneM;
        o1[0]  = fmaxf(acc4[v], 0.0f);
        o1[16] = fmaxf(acc5[v], 0.0f);
        o1[32] = fmaxf(acc6[v], 0.0f);
        o1[48] = fmaxf(acc7[v], 0.0f);
    }
}

extern "C" void kernel_launch(void* const* d_in, const int* in_sizes, int n_in,
                              void* d_out, int out_size, void* d_ws, size_t ws_size,
                              hipStream_t stream) {
    const float*     x  = (const float*)d_in[0];
    const long long* ei = (const long long*)d_in[1];   // [2, E] int64
    const float*     ew = (const float*)d_in[2];
    const float*     W1 = (const float*)d_in[3];
    float* out = (float*)d_out;

    int N = in_sizes[0] / C;
    int E = in_sizes[2];

    // workspace: deg | dinv | agg
    size_t nb = (((size_t)N * sizeof(float)) + 255) & ~(size_t)255;
    float* deg  = (float*)d_ws;
    float* dinv = (float*)((char*)d_ws + nb);
    float* agg  = (float*)((char*)d_ws + 2 * nb);

    const long long* row = ei;        // edge_index[0]
    const long long* col = ei + E;    // edge_index[1]

    init_deg_kernel<<<(N + 255) / 256, 256, 0, stream>>>(deg, N);
    zero_agg_kernel<<<(N * C + 255) / 256, 256, 0, stream>>>(agg, N * C);
    accum_deg_kernel<<<(E + 255) / 256, 256, 0, stream>>>(col, ew, deg, E);
    dinv_kernel<<<(N + 255) / 256, 256, 0, stream>>>(deg, dinv, N);
    scatter_kernel<<<(E + 7) / 8, 256, 0, stream>>>(x, row, col, ew, dinv, agg, E);

    int blk32 = (N + 31) / 32;        // 32-row blocks, one per wave
    gemm_relu_kernel<<<(blk32 + 3) / 4, 128, 0, stream>>>(x, agg, dinv, W1, out, N);
}